// RGAT_8821862826188
// MI455X (gfx1250) — compile-verified
//
#include <hip/hip_runtime.h>
#include <hip/hip_bf16.h>
#include <math.h>

#define N_NODES 50000
#define N_EDGES 800000
#define N_REL   8
#define IN_CH   128
#define HC      256      // HEADS*HID for layers 0/1
#define LN_EPS  1e-5f
#define NEG_SLOPE 0.2f

typedef __attribute__((ext_vector_type(2))) float v2f;
typedef __attribute__((ext_vector_type(8))) float v8f;

static inline int cdiv_l(long long a, long long b) { return (int)((a + b - 1) / b); }

// ---------------------------------------------------------------------------
// Per-relation node transform: XW[r][n][d] = sum_k X[n][k] * W[r][k][d]
// One wave computes a 16x64 output tile (4 x 16x16 WMMA accumulators),
// reusing each A-fragment across 4 column tiles -> 4 WMMAs per A load.
// Nn % 16 == 0, D % 64 == 0; grid sized so every wave is full (EXEC all-1s).
// ---------------------------------------------------------------------------
__global__ __launch_bounds__(256) void k_gemm_rel(
    const float* __restrict__ X,    // [Nn, K]
    const float* __restrict__ W,    // [N_REL, K, D]
    float* __restrict__ XW,         // [N_REL, Nn, D]
    int Nn, int K, int D)
{
    const int lane = threadIdx.x & 31;
    const int wid  = threadIdx.x >> 5;

    const int quadsN = D >> 6;                         // 64-column quads
    const long long tilesPerRel = (long long)(Nn >> 4) * quadsN;
    const long long totalTiles  = tilesPerRel * N_REL;
    const long long gtile = (long long)blockIdx.x * 8 + wid;
    if (gtile >= totalTiles) return;                   // wave-uniform guard

    const int r  = (int)(gtile / tilesPerRel);
    const long long t = gtile % tilesPerRel;
    const int mt = (int)(t / quadsN);                  // 16-row strip
    const int nq = (int)(t % quadsN);                  // 64-col quad

    const int mrow = (mt << 4) + (lane & 15);          // A row for this lane
    const int koff = (lane >> 4) * 2;                  // K sub-offset for this lane
    const int ncol = (nq << 6) + (lane & 15);          // base B/C column for this lane

    const float* __restrict__ Xp = X + (long long)mrow * K + koff;
    const float* __restrict__ Wp = W + ((long long)r * K + koff) * D + ncol;

    v8f acc0 = {}, acc1 = {}, acc2 = {}, acc3 = {};
    for (int k = 0; k < K; k += 4) {
        v2f a;
        a.x = Xp[0];
        a.y = Xp[1];
        v2f b0, b1, b2, b3;
        b0.x = Wp[ 0];      b0.y = Wp[D +  0];
        b1.x = Wp[16];      b1.y = Wp[D + 16];
        b2.x = Wp[32];      b2.y = Wp[D + 32];
        b3.x = Wp[48];      b3.y = Wp[D + 48];
        acc0 = __builtin_amdgcn_wmma_f32_16x16x4_f32(false, a, false, b0, (short)0, acc0, false, false);
        acc1 = __builtin_amdgcn_wmma_f32_16x16x4_f32(false, a, false, b1, (short)0, acc1, false, false);
        acc2 = __builtin_amdgcn_wmma_f32_16x16x4_f32(false, a, false, b2, (short)0, acc2, false, false);
        acc3 = __builtin_amdgcn_wmma_f32_16x16x4_f32(false, a, false, b3, (short)0, acc3, false, false);
        Xp += 4;
        Wp += (long long)4 * D;
    }

    // C/D layout: lane l -> n = l&15 ; VGPR v -> m = v + (l>=16 ? 8 : 0)
    float* __restrict__ Op =
        XW + ((long long)r * Nn + (mt << 4) + ((lane >> 4) << 3)) * D + ncol;
#pragma unroll
    for (int v = 0; v < 8; ++v) {
        long long row = (long long)v * D;
        Op[row +  0] = acc0[v];
        Op[row + 16] = acc1[v];
        Op[row + 32] = acc2[v];
        Op[row + 48] = acc3[v];
    }
}

// ---------------------------------------------------------------------------
// Attention dot precompute: s_src/s_dst[(r*Nn+n)*H + h] = <xw[r,n,h,:], a_src/dst[h,:]>
// ---------------------------------------------------------------------------
__global__ void k_attn_pre(const float* __restrict__ xw,
                           const float* __restrict__ a_src,
                           const float* __restrict__ a_dst,
                           float* __restrict__ s_src,
                           float* __restrict__ s_dst,
                           int Nn, int H, int C)
{
    long long idx = (long long)blockIdx.x * blockDim.x + threadIdx.x;
    long long total = (long long)N_REL * Nn * H;
    if (idx >= total) return;
    int h = (int)(idx % H);
    long long rn = idx / H;                 // r*Nn + n
    const float* xp = xw + rn * ((long long)H * C) + (long long)h * C;
    const float* as = a_src + h * C;
    const float* ad = a_dst + h * C;
    float ss = 0.f, sd = 0.f;
    for (int c = 0; c < C; ++c) {
        float v = xp[c];
        ss = fmaf(v, as[c], ss);
        sd = fmaf(v, ad[c], sd);
    }
    s_src[idx] = ss;
    s_dst[idx] = sd;
}

// monotonic float->uint encoding so unsigned atomicMax == float max
__device__ __forceinline__ unsigned enc_f(float f) {
    unsigned u = __float_as_uint(f);
    return (u & 0x80000000u) ? ~u : (u | 0x80000000u);
}
__device__ __forceinline__ float dec_f(unsigned u) {
    return (u & 0x80000000u) ? __uint_as_float(u & 0x7FFFFFFFu)
                             : __uint_as_float(~u);
}

// ---------------------------------------------------------------------------
// Edge logits + leaky-relu + per-destination running max (encoded atomicMax)
// ---------------------------------------------------------------------------
__global__ void k_edge_logits(const int* __restrict__ srcI,
                              const int* __restrict__ dstI,
                              const int* __restrict__ etI,
                              const float* __restrict__ s_src,
                              const float* __restrict__ s_dst,
                              const float* __restrict__ a_rel,   // [R, H]
                              float* __restrict__ alpha,          // [E, H]
                              unsigned* __restrict__ nmax,        // [Nn, H]
                              int Nn, int H)
{
    long long idx = (long long)blockIdx.x * blockDim.x + threadIdx.x;
    long long total = (long long)N_EDGES * H;
    if (idx >= total) return;
    int h = (int)(idx % H);
    long long e = idx / H;
    int r = etI[e], s = srcI[e], d = dstI[e];
    float a = s_src[((long long)r * Nn + s) * H + h]
            + s_dst[((long long)r * Nn + d) * H + h]
            + a_rel[r * H + h];
    a = (a > 0.f) ? a : NEG_SLOPE * a;
    alpha[idx] = a;
    atomicMax(&nmax[(long long)d * H + h], enc_f(a));
}

// ---------------------------------------------------------------------------
// ex = exp(alpha - max[dst]);  den[dst] += ex   (in-place on alpha buffer)
// ---------------------------------------------------------------------------
__global__ void k_edge_exp(const int* __restrict__ dstI,
                           float* __restrict__ alpha,        // in/out [E, H]
                           const unsigned* __restrict__ nmax,
                           float* __restrict__ den,          // [Nn, H]
                           int H)
{
    long long idx = (long long)blockIdx.x * blockDim.x + threadIdx.x;
    long long total = (long long)N_EDGES * H;
    if (idx >= total) return;
    int h = (int)(idx % H);
    long long e = idx / H;
    int d = dstI[e];
    float m = dec_f(nmax[(long long)d * H + h]);
    float v = expf(alpha[idx] - m);
    alpha[idx] = v;
    atomicAdd(&den[(long long)d * H + h], v);
}

// ---------------------------------------------------------------------------
// Message scatter: out[dst][d] += xw[et][src][d] * ex[e][h]/den[dst][h]
// One thread per (edge, channel); D=256 -> one block per edge (coalesced row).
// ---------------------------------------------------------------------------
__global__ void k_edge_scatter(const int* __restrict__ srcI,
                               const int* __restrict__ dstI,
                               const int* __restrict__ etI,
                               const float* __restrict__ xw,   // [R, Nn, D]
                               const float* __restrict__ ex,   // [E, H]
                               const float* __restrict__ den,  // [Nn, H]
                               float* __restrict__ out,        // [Nn, D]
                               int Nn, int H, int C)
{
    const int D = H * C;
    long long idx = (long long)blockIdx.x * blockDim.x + threadIdx.x;
    long long total = (long long)N_EDGES * D;
    if (idx >= total) return;
    int dch = (int)(idx % D);
    long long e = idx / D;
    int h = dch / C;
    int r = etI[e], s = srcI[e], dn = dstI[e];
    float w = ex[e * H + h] / den[(long long)dn * H + h];
    float v = xw[((long long)r * Nn + s) * D + dch] * w;
    atomicAdd(&out[(long long)dn * D + dch], v);
}

// ---------------------------------------------------------------------------
// h = elu(layernorm(acc + bias))  — one wave (32 lanes) per node, D = 256
// ---------------------------------------------------------------------------
__global__ __launch_bounds__(256) void k_bias_ln_elu(
    const float* __restrict__ acc, const float* __restrict__ bias,
    const float* __restrict__ g,   const float* __restrict__ be,
    float* __restrict__ out, int Nn)
{
    const int D = HC;                      // 256
    const int lane = threadIdx.x & 31;
    const int node = blockIdx.x * 8 + (threadIdx.x >> 5);
    if (node >= Nn) return;
    const float* p = acc + (long long)node * D;

    float vals[8];
    float s = 0.f;
#pragma unroll
    for (int i = 0; i < 8; ++i) {
        int c = lane + i * 32;
        float v = p[c] + bias[c];
        vals[i] = v;
        s += v;
    }
#pragma unroll
    for (int o = 16; o > 0; o >>= 1) s += __shfl_xor(s, o, 32);
    float mean = s / (float)D;

    float vs = 0.f;
#pragma unroll
    for (int i = 0; i < 8; ++i) { float d0 = vals[i] - mean; vs += d0 * d0; }
#pragma unroll
    for (int o = 16; o > 0; o >>= 1) vs += __shfl_xor(vs, o, 32);
    float inv = 1.0f / sqrtf(vs / (float)D + LN_EPS);

    float* op = out + (long long)node * D;
#pragma unroll
    for (int i = 0; i < 8; ++i) {
        int c = lane + i * 32;
        float y = g[c] * (vals[i] - mean) * inv + be[c];
        op[c] = (y > 0.f) ? y : (expm1f(y));   // ELU(alpha=1)
    }
}

__global__ void k_bias_add(float* __restrict__ out, const float* __restrict__ bias,
                           int Nn, int D)
{
    long long idx = (long long)blockIdx.x * blockDim.x + threadIdx.x;
    if (idx >= (long long)Nn * D) return;
    out[idx] += bias[idx % D];
}

// ---------------------------------------------------------------------------
// Host-side layer driver (all launches on `stream`, graph-capture safe)
// ---------------------------------------------------------------------------
static void run_rgat_layer(const float* X, int K,
                           const float* W, const float* a_s, const float* a_d,
                           const float* a_r, int H, int C,
                           const int* srcI, const int* dstI, const int* etI,
                           float* xw, float* ssrc, float* sdst, float* ex,
                           unsigned* nmax, float* den, float* outAcc,
                           hipStream_t stream)
{
    const int Nn = N_NODES;
    const int D = H * C;

    long long tiles = (long long)N_REL * (Nn >> 4) * (D >> 6);
    k_gemm_rel<<<cdiv_l(tiles, 8), 256, 0, stream>>>(X, W, xw, Nn, K, D);

    long long pre = (long long)N_REL * Nn * H;
    k_attn_pre<<<cdiv_l(pre, 256), 256, 0, stream>>>(xw, a_s, a_d, ssrc, sdst, Nn, H, C);

    hipMemsetAsync(nmax, 0, sizeof(unsigned) * (size_t)Nn * H, stream);
    hipMemsetAsync(den,  0, sizeof(float)    * (size_t)Nn * H, stream);
    hipMemsetAsync(outAcc, 0, sizeof(float) * (size_t)Nn * D, stream);

    long long eh = (long long)N_EDGES * H;
    k_edge_logits<<<cdiv_l(eh, 256), 256, 0, stream>>>(srcI, dstI, etI, ssrc, sdst,
                                                       a_r, ex, nmax, Nn, H);
    k_edge_exp<<<cdiv_l(eh, 256), 256, 0, stream>>>(dstI, ex, nmax, den, H);

    long long ed = (long long)N_EDGES * D;
    k_edge_scatter<<<cdiv_l(ed, 256), 256, 0, stream>>>(srcI, dstI, etI, xw, ex, den,
                                                        outAcc, Nn, H, C);
}

extern "C" void kernel_launch(void* const* d_in, const int* in_sizes, int n_in,
                              void* d_out, int out_size, void* d_ws, size_t ws_size,
                              hipStream_t stream)
{
    (void)in_sizes; (void)n_in; (void)out_size; (void)ws_size;
    const float* x   = (const float*)d_in[0];
    const int*   ei  = (const int*)d_in[1];
    const int*   et  = (const int*)d_in[2];
    const float* W0  = (const float*)d_in[3];
    const float* as0 = (const float*)d_in[4];
    const float* ad0 = (const float*)d_in[5];
    const float* ar0 = (const float*)d_in[6];
    const float* bi0 = (const float*)d_in[7];
    const float* W1  = (const float*)d_in[8];
    const float* as1 = (const float*)d_in[9];
    const float* ad1 = (const float*)d_in[10];
    const float* ar1 = (const float*)d_in[11];
    const float* bi1 = (const float*)d_in[12];
    const float* W2  = (const float*)d_in[13];
    const float* as2 = (const float*)d_in[14];
    const float* ad2 = (const float*)d_in[15];
    const float* ar2 = (const float*)d_in[16];
    const float* bi2 = (const float*)d_in[17];
    const float* g0  = (const float*)d_in[18];
    const float* be0 = (const float*)d_in[19];
    const float* g1  = (const float*)d_in[20];
    const float* be1 = (const float*)d_in[21];

    const int* srcI = ei;               // edge_index[0]
    const int* dstI = ei + N_EDGES;     // edge_index[1]

    // workspace partition (floats)
    float* xw   = (float*)d_ws;                                  // R*Nn*256
    float* h    = xw   + (size_t)N_REL * N_NODES * HC;           // Nn*256
    float* acc  = h    + (size_t)N_NODES * HC;                   // Nn*256
    float* ssrc = acc  + (size_t)N_NODES * HC;                   // R*Nn*4
    float* sdst = ssrc + (size_t)N_REL * N_NODES * 4;            // R*Nn*4
    float* ex   = sdst + (size_t)N_REL * N_NODES * 4;            // E*4
    unsigned* nmax = (unsigned*)(ex + (size_t)N_EDGES * 4);      // Nn*4
    float* den  = (float*)(nmax + (size_t)N_NODES * 4);          // Nn*4

    // Layer 0: 128 -> (4 heads x 64), concat; then LN + ELU
    run_rgat_layer(x, IN_CH, W0, as0, ad0, ar0, 4, 64,
                   srcI, dstI, et, xw, ssrc, sdst, ex, nmax, den, acc, stream);
    k_bias_ln_elu<<<cdiv_l(N_NODES, 8), 256, 0, stream>>>(acc, bi0, g0, be0, h, N_NODES);

    // Layer 1: 256 -> (4 heads x 64), concat; then LN + ELU
    run_rgat_layer(h, HC, W1, as1, ad1, ar1, 4, 64,
                   srcI, dstI, et, xw, ssrc, sdst, ex, nmax, den, acc, stream);
    k_bias_ln_elu<<<cdiv_l(N_NODES, 8), 256, 0, stream>>>(acc, bi1, g1, be1, h, N_NODES);

    // Layer 2: 256 -> (1 head x 64), mean over 1 head == identity; accumulate into d_out
    run_rgat_layer(h, HC, W2, as2, ad2, ar2, 1, 64,
                   srcI, dstI, et, xw, ssrc, sdst, ex, nmax, den,
                   (float*)d_out, stream);
    k_bias_add<<<cdiv_l((long long)N_NODES * 64, 256), 256, 0, stream>>>(
        (float*)d_out, bi2, N_NODES, 64);
}